// RPNPostProcessor_74543452389407
// MI455X (gfx1250) — compile-verified
//
#include <hip/hip_runtime.h>
#include <stdint.h>
#include <math.h>

#define NB    4          // batches
#define NA    262144     // anchors per batch
#define NTOT  (NB * NA)
#define PRE   4000
#define POST  1000
#define CAND  8192
#define NBIN  2048
#define NWRD  125        // 4000 bits = 125 u32 words
#define RPAD  128        // row stride in words (512B, b128-friendly)
#define NMS_T 0.3f
#define AUGT  0.2f

#if defined(__has_builtin)
#if __has_builtin(__builtin_amdgcn_tensor_load_to_lds)
#define USE_TDM 1
#endif
#endif

typedef unsigned int  uint32x4_t __attribute__((ext_vector_type(4)));
typedef int           int32x8_t  __attribute__((ext_vector_type(8)));
typedef int           int32x4_t  __attribute__((ext_vector_type(4)));

// Order-preserving map: larger float -> larger key (total order incl. sign).
__device__ __forceinline__ unsigned int fkey(float f) {
    unsigned int u = __float_as_uint(f);
    return (u & 0x80000000u) ? ~u : (u | 0x80000000u);
}

// ---------------------------------------------------------------- zero init
__global__ void zero_kernel(unsigned int* p, int n) {
    int i = blockIdx.x * blockDim.x + threadIdx.x;
    if (i < n) p[i] = 0u;
}

// ---------------------------------------------------------------- histogram
__global__ void hist_kernel(const float* __restrict__ obj, unsigned int* __restrict__ hist) {
    __shared__ unsigned int h[NBIN];
    const int tid  = threadIdx.x;
    const int base = blockIdx.x * 1024;
    const int b    = base / NA;
    for (int i = tid; i < NBIN; i += 256) h[i] = 0u;
    __syncthreads();
#pragma unroll
    for (int e = 0; e < 4; ++e) {
        int g = base + e * 256 + tid;
        __builtin_prefetch(obj + ((g + 8192) & (NTOT - 1)), 0, 0);  // global_prefetch_b8
        unsigned int key = fkey(obj[g]);
        atomicAdd(&h[key >> 21], 1u);
    }
    __syncthreads();
    for (int i = tid; i < NBIN; i += 256)
        if (h[i]) atomicAdd(&hist[b * NBIN + i], h[i]);
}

// ------------------------------------------------------- threshold bin scan
__global__ void thr_kernel(const unsigned int* __restrict__ hist, unsigned int* __restrict__ thr) {
    if (threadIdx.x != 0) return;
    int b = blockIdx.x;
    unsigned int c = 0, t = 0;
    for (int bin = NBIN - 1; bin >= 0; --bin) {
        c += hist[b * NBIN + bin];
        if (c >= PRE) { t = (unsigned int)bin; break; }
    }
    thr[b] = t;
}

// ---------------------------------------------------------------- compaction
__global__ void compact_kernel(const float* __restrict__ obj, const unsigned int* __restrict__ thr,
                               unsigned int* __restrict__ cnt, unsigned int* __restrict__ ckey,
                               unsigned int* __restrict__ cidx) {
    const int tid  = threadIdx.x;
    const int base = blockIdx.x * 1024;
    const int b    = base / NA;
    const unsigned int t = thr[b];
#pragma unroll
    for (int e = 0; e < 4; ++e) {
        int g = base + e * 256 + tid;
        unsigned int key = fkey(obj[g]);
        if ((key >> 21) >= t) {
            unsigned int p = atomicAdd(&cnt[b], 1u);
            if (p < CAND) {
                ckey[b * CAND + p] = key;
                cidx[b * CAND + p] = (unsigned int)(g - b * NA);
            }
        }
    }
}

// ------------------------------------- bitonic sort (key desc, idx asc) + top-PRE
__global__ __launch_bounds__(1024)
void sort_kernel(const float* __restrict__ obj, const unsigned int* __restrict__ cnt,
                 const unsigned int* __restrict__ ckey, const unsigned int* __restrict__ cidx,
                 unsigned int* __restrict__ tidx, float* __restrict__ tscore) {
    __shared__ unsigned int sk[CAND];
    __shared__ unsigned int si[CAND];
    const int b   = blockIdx.x;
    const int tid = threadIdx.x;
    unsigned int n = cnt[b]; if (n > CAND) n = CAND;
#pragma unroll
    for (int e = 0; e < 8; ++e) {
        unsigned int i = (unsigned)(e * 1024 + tid);
        if (i < n) { sk[i] = ckey[b * CAND + i]; si[i] = cidx[b * CAND + i]; }
        else       { sk[i] = 0u;                 si[i] = 0xFFFFFFFFu; }
    }
    for (unsigned int k = 2; k <= CAND; k <<= 1) {
        for (unsigned int j = k >> 1; j > 0; j >>= 1) {
            __syncthreads();
#pragma unroll
            for (int e = 0; e < 8; ++e) {
                unsigned int i   = (unsigned)(e * 1024 + tid);
                unsigned int ixj = i ^ j;
                if (ixj > i) {
                    unsigned int ka = sk[i], kb = sk[ixj];
                    unsigned int ia = si[i], ib = si[ixj];
                    bool up = ((i & k) == 0);
                    bool b_before_a = (kb > ka) || (kb == ka && ib < ia);
                    bool a_before_b = (ka > kb) || (ka == kb && ia < ib);
                    if (up ? b_before_a : a_before_b) {
                        sk[i] = kb; sk[ixj] = ka;
                        si[i] = ib; si[ixj] = ia;
                    }
                }
            }
        }
    }
    __syncthreads();
#pragma unroll
    for (int e = 0; e < 4; ++e) {
        int r = e * 1024 + tid;
        if (r < PRE) {
            unsigned int idx = si[r];
            if (idx == 0xFFFFFFFFu) idx = 0;  // cannot happen if >=PRE candidates
            tidx[b * PRE + r] = idx;
            float lg = obj[(size_t)b * NA + idx];
            tscore[b * PRE + r] = 1.0f / (1.0f + expf(-lg));
        }
    }
}

// ------------------------------------------------------ decode + aug AABB SoA
__global__ void decode_kernel(const float* __restrict__ reg, const float* __restrict__ anc,
                              const unsigned int* __restrict__ tidx,
                              float* __restrict__ boxes, float* __restrict__ aug) {
    const int b = blockIdx.y;
    const int r = blockIdx.x * 256 + threadIdx.x;
    if (r >= PRE) return;
    unsigned int i = tidx[b * PRE + r];
    size_t o = ((size_t)b * NA + i) * 7;
    float xa = anc[o], ya = anc[o+1], za = anc[o+2], wa = anc[o+3],
          la = anc[o+4], ha = anc[o+5], ra = anc[o+6];
    float dx = reg[o], dy = reg[o+1], dz = reg[o+2], dw = reg[o+3],
          dl = reg[o+4], dh = reg[o+5], dr = reg[o+6];
    float diag = sqrtf(wa * wa + la * la);
    float x = dx * diag + xa;
    float y = dy * diag + ya;
    float z = dz * ha + za;
    float w = expf(dw) * wa;
    float l = expf(dl) * la;
    float h = expf(dh) * ha;
    float rr = dr + ra;
    size_t bo = ((size_t)b * PRE + r) * 7;
    boxes[bo+0] = x; boxes[bo+1] = y; boxes[bo+2] = z;
    boxes[bo+3] = w; boxes[bo+4] = l; boxes[bo+5] = h; boxes[bo+6] = rr;
    float sx = fmaxf(w, AUGT), sy = fmaxf(l, AUGT), sz = fmaxf(h, AUGT);
    size_t ab = (size_t)b * 7 * PRE;
    aug[ab + 0 * PRE + r] = x - 0.5f * sx;     // lo.x
    aug[ab + 1 * PRE + r] = y - 0.5f * sy;     // lo.y
    aug[ab + 2 * PRE + r] = z;                 // lo.z
    aug[ab + 3 * PRE + r] = x + 0.5f * sx;     // hi.x
    aug[ab + 4 * PRE + r] = y + 0.5f * sy;     // hi.y
    aug[ab + 5 * PRE + r] = z + sz;            // hi.z
    aug[ab + 6 * PRE + r] = sx * sy * sz;      // volume
}

// ------------------- phase 1: pairwise overlap bit-matrix (TDM-staged LDS table)
__global__ __launch_bounds__(1024)
void nms_bitmat_kernel(const float* __restrict__ aug, unsigned int* __restrict__ bitmat) {
    __shared__ float s[7 * PRE];               // 112 KB box table in CDNA5 LDS
    const int b = blockIdx.y, tid = threadIdx.x;
    const float* src = aug + (size_t)b * 7 * PRE;

#ifdef USE_TDM
    if (tid < 32) {  // wave 0 issues one TDM descriptor: 28000 contiguous dwords -> LDS
        unsigned int lds = (unsigned int)(uintptr_t)(void*)s;   // low 32b of flat = LDS offset
        unsigned long long ga = (unsigned long long)(uintptr_t)src;
        const unsigned int nd0 = 7u * PRE;                      // 28000 elements, 4B each
        uint32x4_t g0;
        g0.x = 1u;                                              // count=1, user descriptor
        g0.y = lds;                                             // lds_addr
        g0.z = (unsigned int)ga;                                // global_addr[31:0]
        g0.w = (unsigned int)((ga >> 32) & 0x1FFFFFFu) | (2u << 30);  // ga[56:32] | type=2
        int32x8_t g1;
        g1[0] = (int)(2u << 16);                                // data_size=4B
        g1[1] = (int)((nd0 & 0xFFFFu) << 16);                   // tensor_dim0[15:0]
        g1[2] = (int)((nd0 >> 16) | (1u << 16));                // tensor_dim0[31:16], tensor_dim1=1
        g1[3] = (int)((nd0 & 0xFFFFu) << 16);                   // tile_dim0 = 28000
        g1[4] = (int)1;                                         // tile_dim1=1, tile_dim2=0
        g1[5] = (int)nd0;                                       // tensor_dim0_stride
        g1[6] = 0;
        g1[7] = 0;
        int32x4_t g2 = {0, 0, 0, 0};
        int32x4_t g3 = {0, 0, 0, 0};
        int32x8_t g4 = {0, 0, 0, 0, 0, 0, 0, 0};
        __builtin_amdgcn_tensor_load_to_lds(g0, g1, g2, g3, g4, 0);
        __builtin_amdgcn_s_wait_tensorcnt(0);
    }
#else
    for (int i = tid; i < 7 * PRE; i += 1024) s[i] = src[i];
#endif
    __syncthreads();

    const float* lox = s;            const float* loy = s + PRE;
    const float* loz = s + 2 * PRE;  const float* hix = s + 3 * PRE;
    const float* hiy = s + 4 * PRE;  const float* hiz = s + 5 * PRE;
    const float* vol = s + 6 * PRE;

    const int wv = tid >> 5, lane = tid & 31;
    const int i = blockIdx.x * 32 + wv;        // one wave per row; 125*32 == 4000
    float bx0 = lox[i], by0 = loy[i], bz0 = loz[i];
    float bx1 = hix[i], by1 = hiy[i], bz1 = hiz[i], bv = vol[i];
    unsigned int* row = bitmat + ((size_t)b * PRE + i) * RPAD;

    for (int w2 = 0; w2 < NWRD; ++w2) {
        int j = w2 * 32 + lane;                // always < 4000
        float ix = fminf(bx1, hix[j]) - fmaxf(bx0, lox[j]);
        float iy = fminf(by1, hiy[j]) - fmaxf(by0, loy[j]);
        float iz = fminf(bz1, hiz[j]) - fmaxf(bz0, loz[j]);
        float inter = fmaxf(ix, 0.f) * fmaxf(iy, 0.f) * fmaxf(iz, 0.f);
        float uni = bv + vol[j] - inter;
        bool pred = (j != i) && (inter > NMS_T * fmaxf(uni, 1e-8f));
        unsigned int word = (unsigned int)__ballot(pred);   // wave32: low 32 bits
        if (lane == (w2 & 31)) row[w2] = word;
    }
    if (lane < RPAD - NWRD) row[NWRD + lane] = 0u;          // zero padding words
}

// ------------------- phase 2: single-wave serial scan, mask in registers
__global__ __launch_bounds__(32)
void nms_scan_kernel(const unsigned int* __restrict__ bitmat, unsigned int* __restrict__ keepw) {
    const int b = blockIdx.x, lane = threadIdx.x;
    unsigned int m0 = 0, m1 = 0, m2 = 0, m3 = 0;            // lane owns words 4*lane..4*lane+3
    const uint4* rp = (const uint4*)(bitmat + (size_t)b * PRE * RPAD);
    uint4 nxt = rp[lane];                                   // row 0
    for (int i = 0; i < PRE; ++i) {
        uint4 cur = nxt;
        if (i + 1 < PRE) nxt = rp[(size_t)(i + 1) * 32 + lane];  // speculative prefetch
        int wi = i >> 5;
        int owner = wi >> 2, sub = wi & 3;                  // uniform across lanes
        unsigned int myw = (sub == 0) ? m0 : (sub == 1) ? m1 : (sub == 2) ? m2 : m3;
        unsigned int v = __shfl(myw, owner, 32);
        if (((v >> (i & 31)) & 1u) == 0u) {                 // pivot i kept -> OR its row
            m0 |= cur.x; m1 |= cur.y; m2 |= cur.z; m3 |= cur.w;
        }
    }
    unsigned int* kp = keepw + b * RPAD + lane * 4;         // keep = ~suppressed
    kp[0] = ~m0; kp[1] = ~m1; kp[2] = ~m2; kp[3] = ~m3;
}

// -------------------------------------------------- compact kept -> (B,1000,8)
__global__ __launch_bounds__(1024)
void finalize_kernel(const float* __restrict__ boxes, const float* __restrict__ tscore,
                     const unsigned int* __restrict__ keepw, float* __restrict__ out) {
    __shared__ unsigned char  k[PRE];
    __shared__ unsigned short p[PRE];
    const int b = blockIdx.x, tid = threadIdx.x;
    for (int r = tid; r < PRE; r += 1024)
        k[r] = (unsigned char)((keepw[b * RPAD + (r >> 5)] >> (r & 31)) & 1u);
    __syncthreads();
    if (tid == 0) {
        unsigned int c = 0;
        for (int r = 0; r < PRE; ++r) { p[r] = (unsigned short)c; c += k[r]; }
    }
    for (int sidx = tid; sidx < POST * 8; sidx += 1024)
        out[(size_t)b * POST * 8 + sidx] = 0.0f;
    __syncthreads();
    for (int r = tid; r < PRE; r += 1024) {
        if (k[r]) {
            unsigned int pos = p[r];
            if (pos < POST) {
                size_t ob = ((size_t)b * POST + pos) * 8;
                size_t bo = ((size_t)b * PRE + r) * 7;
#pragma unroll
                for (int c = 0; c < 7; ++c) out[ob + c] = boxes[bo + c];
                out[ob + 7] = tscore[b * PRE + r];
            }
        }
    }
}

extern "C" void kernel_launch(void* const* d_in, const int* in_sizes, int n_in,
                              void* d_out, int out_size, void* d_ws, size_t ws_size,
                              hipStream_t stream) {
    const float* obj = (const float*)d_in[0];
    const float* reg = (const float*)d_in[1];
    const float* anc = (const float*)d_in[2];
    float* out = (float*)d_out;

    unsigned int* ws     = (unsigned int*)d_ws;
    unsigned int* hist   = ws;                       // B*NBIN
    unsigned int* thr    = hist + NB * NBIN;         // B
    unsigned int* cnt    = thr + NB;                 // B
    unsigned int* ckey   = cnt + NB;                 // B*CAND
    unsigned int* cidx   = ckey + NB * CAND;         // B*CAND
    unsigned int* tidx   = cidx + NB * CAND;         // B*PRE
    float*        tscore = (float*)(tidx + NB * PRE);    // B*PRE
    float*        boxes  = tscore + NB * PRE;             // B*PRE*7
    float*        aug    = boxes + (size_t)NB * PRE * 7;  // B*7*PRE
    unsigned int* keepw  = (unsigned int*)(aug + (size_t)NB * 7 * PRE); // B*RPAD
    unsigned int* bitmat = keepw + NB * RPAD;        // B*PRE*RPAD (~8 MB, L2-resident)

    int nz = NB * NBIN + 2 * NB;
    zero_kernel<<<(nz + 255) / 256, 256, 0, stream>>>(ws, nz);
    hist_kernel<<<NTOT / 1024, 256, 0, stream>>>(obj, hist);
    thr_kernel<<<NB, 32, 0, stream>>>(hist, thr);
    compact_kernel<<<NTOT / 1024, 256, 0, stream>>>(obj, thr, cnt, ckey, cidx);
    sort_kernel<<<NB, 1024, 0, stream>>>(obj, cnt, ckey, cidx, tidx, tscore);
    decode_kernel<<<dim3((PRE + 255) / 256, NB), 256, 0, stream>>>(reg, anc, tidx, boxes, aug);
    nms_bitmat_kernel<<<dim3(PRE / 32, NB), 1024, 0, stream>>>(aug, bitmat);
    nms_scan_kernel<<<NB, 32, 0, stream>>>(bitmat, keepw);
    finalize_kernel<<<NB, 1024, 0, stream>>>(boxes, tscore, keepw, out);
}